// FastHyperbolicMultiheadAttention_5952824672947
// MI455X (gfx1250) — compile-verified
//
#include <hip/hip_runtime.h>

// ---------------------------------------------------------------------------
// FastHyperbolicMultiheadAttention for MI455X (gfx1250, wave32, WMMA)
// fp32 end-to-end using V_WMMA_F32_16X16X4_F32 (exact fp32 semantics).
// Tile staging via GLOBAL_LOAD_ASYNC_TO_LDS (ASYNCcnt-tracked).
// ---------------------------------------------------------------------------

typedef __attribute__((ext_vector_type(2))) float v2f;
typedef __attribute__((ext_vector_type(8))) float v8f;
typedef __attribute__((ext_vector_type(4))) int   v4i;

#define EDIM 512
#define SDIM 2048
#define BDIM 4
#define HDIM 8
#define DH   64
#define MROWS (BDIM * SDIM)   // 8192 token rows

// ---- capability probes (answers show up in compiler stderr) ---------------
#if __has_builtin(__builtin_amdgcn_global_load_async_to_lds_b128)
#define HAVE_ASYNC_LDS 1
#else
#define HAVE_ASYNC_LDS 0
#warning "probe: NO __builtin_amdgcn_global_load_async_to_lds_b128"
#endif
#if !__has_builtin(__builtin_amdgcn_s_wait_asynccnt)
#warning "probe: NO __builtin_amdgcn_s_wait_asynccnt"
#endif

// address-space-qualified 16-byte vector pointers (per builtin signature:
// 'int __vector(4) __device__ *' for the global side)
typedef __attribute__((address_space(1))) v4i g_v4i;
typedef __attribute__((address_space(3))) v4i l_v4i;

// 16-byte global -> LDS tile staging
static __device__ __forceinline__ void stage16(float* lds_dst,
                                               const float* gsrc) {
#if HAVE_ASYNC_LDS
  // generic LDS pointer: low 32 bits are the LDS byte offset;
  // generic global pointer == AS1 address numerically.
  __builtin_amdgcn_global_load_async_to_lds_b128(
      (g_v4i*)(uintptr_t)gsrc, (l_v4i*)(uintptr_t)lds_dst, 0, 0);
#else
  *(float4*)lds_dst = *(const float4*)gsrc;
#endif
}

static __device__ __forceinline__ void stage_wait() {
#if HAVE_ASYNC_LDS
#if __has_builtin(__builtin_amdgcn_s_wait_asynccnt)
  __builtin_amdgcn_s_wait_asynccnt(0);
#else
  asm volatile("s_wait_asynccnt 0x0" ::: "memory");
#endif
#endif
}

static __device__ __forceinline__ v8f wmma_f32(v2f a, v2f b, v8f c) {
  // 8 args: (neg_a, A, neg_b, B, c_mod, C, reuse_a, reuse_b)
  return __builtin_amdgcn_wmma_f32_16x16x4_f32(false, a, false, b, (short)0, c,
                                               false, false);
}

static __device__ __forceinline__ float wave_sum(float v) {
#pragma unroll
  for (int d = 1; d < 32; d <<= 1) v += __shfl_xor(v, d, 32);
  return v;
}

// ---------------------------------------------------------------------------
// GEMM (NT): C[M,512] = A[M,512] * W[512,512]^T, fp32, WMMA 16x16x4.
// Block: 128 threads (4 waves), 64x64 output tile, K staged via LDS (32-chunk).
// Wave w covers cols [16w,16w+16); 4 stacked 16x16 M-subtiles.
// Fragments for a whole K-chunk are preloaded, then 32 WMMAs issue densely.
// ---------------------------------------------------------------------------
__global__ __launch_bounds__(128) void gemm_nt_kernel(
    const float* __restrict__ A, const float* __restrict__ W,
    float* __restrict__ C) {
  __shared__ alignas(16) float As[64][36];
  __shared__ alignas(16) float Bs[64][36];

  const int tid  = threadIdx.x;
  const int lane = tid & 31;
  const int wv   = tid >> 5;               // 0..3
  const int m0   = blockIdx.y * 64;
  const int n0   = blockIdx.x * 64;

  v8f acc[4];
#pragma unroll
  for (int s = 0; s < 4; ++s) acc[s] = (v8f){0.f,0.f,0.f,0.f,0.f,0.f,0.f,0.f};

  const int lrow  = tid >> 3;              // 0..15 (8 threads per row)
  const int lcol4 = (tid & 7) * 4;         // 0..28
  const int ml    = lane & 15;
  const int kh    = (lane >> 4) * 2;       // 0 or 2 (A/B frag K halves)

  for (int k0 = 0; k0 < EDIM; k0 += 32) {
#pragma unroll
    for (int rr = 0; rr < 64; rr += 16) {
      stage16(&As[rr + lrow][lcol4],
              &A[(size_t)(m0 + rr + lrow) * EDIM + k0 + lcol4]);
      stage16(&Bs[rr + lrow][lcol4],
              &W[(size_t)(n0 + rr + lrow) * EDIM + k0 + lcol4]);
    }
    stage_wait();
    __syncthreads();

    v2f bf[8], af[4][8];
#pragma unroll
    for (int k8 = 0; k8 < 8; ++k8) {
      const int kk = 4 * k8 + kh;
      bf[k8].x = Bs[wv * 16 + ml][kk];
      bf[k8].y = Bs[wv * 16 + ml][kk + 1];
#pragma unroll
      for (int s = 0; s < 4; ++s) {
        af[s][k8].x = As[s * 16 + ml][kk];
        af[s][k8].y = As[s * 16 + ml][kk + 1];
      }
    }
#pragma unroll
    for (int k8 = 0; k8 < 8; ++k8)
#pragma unroll
      for (int s = 0; s < 4; ++s)
        acc[s] = wmma_f32(af[s][k8], bf[k8], acc[s]);
    __syncthreads();
  }

  const int colg  = n0 + wv * 16 + ml;
  const int rbase = (lane >> 4) * 8;
#pragma unroll
  for (int s = 0; s < 4; ++s)
#pragma unroll
    for (int i = 0; i < 8; ++i)
      C[(size_t)(m0 + s * 16 + rbase + i) * EDIM + colg] = acc[s][i];
}

// ---------------------------------------------------------------------------
// Hyperbolic pointwise: out = project(mobius_add(project(scale*Mx), bias))
// One wave per token row (512 elems = 16 per lane). In-place safe.
// ---------------------------------------------------------------------------
__global__ __launch_bounds__(256) void hyp_pointwise_kernel(
    const float* __restrict__ X, const float* MX, const float* __restrict__ bias,
    float* out) {
  const int row  = blockIdx.x * (blockDim.x >> 5) + (threadIdx.x >> 5);
  const int lane = threadIdx.x & 31;
  const size_t base = (size_t)row * EDIM;

  float xr[16], mxr[16], br[16];
  float xn2 = 0.f, mxn2 = 0.f;
#pragma unroll
  for (int j = 0; j < 16; ++j) {
    const int idx = lane + 32 * j;
    xr[j]  = X[base + idx];
    mxr[j] = MX[base + idx];
    br[j]  = bias[idx];
    xn2  += xr[j] * xr[j];
    mxn2 += mxr[j] * mxr[j];
  }
  xn2  = wave_sum(xn2);
  mxn2 = wave_sum(mxn2);

  const float MINN = 1e-15f;
  const float MAXN = 1.0f - 4e-3f;     // (1 - BALL_EPS) / sqrt(c), c = 1
  float xnorm  = fmaxf(sqrtf(xn2), MINN);
  float mxnorm = fmaxf(sqrtf(mxn2), MINN);
  float t  = fminf(xnorm, 1.0f - 1e-7f);            // artanh clip (xnorm >= 0)
  float ar = 0.5f * (log1pf(t) - log1pf(-t));
  float s  = tanhf(mxnorm / xnorm * ar) / mxnorm;   // res = s * Mx
  float resn = s * mxnorm;                          // = tanh(...) >= 0
  if (resn > MAXN) s *= MAXN / resn;                // project()

  // mobius_add(res, bias)
  float x2 = 0.f, y2 = 0.f, xy = 0.f;
#pragma unroll
  for (int j = 0; j < 16; ++j) {
    float r = s * mxr[j];
    x2 += r * r;
    y2 += br[j] * br[j];
    xy += r * br[j];
  }
  x2 = wave_sum(x2);
  y2 = wave_sum(y2);
  xy = wave_sum(xy);

  float c1  = 1.0f + 2.0f * xy + y2;                // coeff on res
  float c2  = 1.0f - x2;                            // coeff on bias
  float den = fmaxf(1.0f + 2.0f * xy + x2 * y2, MINN);
  float inv = 1.0f / den;

  float o[16];
  float on2 = 0.f;
#pragma unroll
  for (int j = 0; j < 16; ++j) {
    o[j] = (c1 * (s * mxr[j]) + c2 * br[j]) * inv;
    on2 += o[j] * o[j];
  }
  on2 = wave_sum(on2);
  float onorm = fmaxf(sqrtf(on2), MINN);
  float ps = (onorm > MAXN) ? (MAXN / onorm) : 1.0f;
#pragma unroll
  for (int j = 0; j < 16; ++j) out[base + lane + 32 * j] = o[j] * ps;
}

// ---------------------------------------------------------------------------
// Per-(row, head) squared norms. Head slice is contiguous: pair w = row*H + h
// lives at buf + w*64. One wave per pair (2 elems/lane).
// ---------------------------------------------------------------------------
__global__ __launch_bounds__(256) void head_norms_kernel(
    const float* __restrict__ buf, float* __restrict__ n2) {
  const int w    = blockIdx.x * (blockDim.x >> 5) + (threadIdx.x >> 5);
  const int lane = threadIdx.x & 31;
  const size_t base = (size_t)w * DH;
  float a = buf[base + lane];
  float b = buf[base + 32 + lane];
  float s = wave_sum(a * a + b * b);
  if (lane == 0) n2[w] = s;
}

// ---------------------------------------------------------------------------
// Flash-style hyperbolic attention. Block = 256 threads (8 waves, 4(M)x2(N)),
// one (b, h, 64-query tile) per block; iterate 32 key tiles of 64.
// QK^T and P*V both via fp32 WMMA; online softmax state in LDS.
// ---------------------------------------------------------------------------
__global__ __launch_bounds__(256) void flash_attn_kernel(
    const float* __restrict__ Q, const float* __restrict__ K,
    const float* __restrict__ V, const float* __restrict__ QN2,
    const float* __restrict__ KN2, float* __restrict__ O) {
  const int qt = blockIdx.x;     // query tile  0..31
  const int hh = blockIdx.y;     // head        0..7
  const int bb = blockIdx.z;     // batch       0..3
  const int tid  = threadIdx.x;
  const int lane = tid & 31;
  const int wv   = tid >> 5;
  const int wm   = wv >> 1;      // 0..3 : query-row band
  const int wn   = wv & 1;       // 0..1 : 32-col band

  __shared__ alignas(16) float Qs[64][68];
  __shared__ alignas(16) float Ks[64][68];
  __shared__ alignas(16) float Vs[64][68];
  __shared__ alignas(16) float Ps[64][68];
  __shared__ float qn2s[64], kn2s[64];
  __shared__ float m_i[64], l_i[64], fsc[64];
  __shared__ float pred[2][64];

  const size_t rowbase = ((size_t)bb * SDIM + qt * 64) * EDIM + hh * DH;
  const int r  = tid >> 4;          // 0..15
  const int c4 = (tid & 15) * 4;    // 0..60

#pragma unroll
  for (int rr = 0; rr < 64; rr += 16)
    stage16(&Qs[rr + r][c4], &Q[rowbase + (size_t)(rr + r) * EDIM + c4]);
  if (tid < 64) {
    qn2s[tid] = fminf(QN2[((size_t)bb * SDIM + qt * 64 + tid) * HDIM + hh],
                      1.0f - 1e-5f);
    m_i[tid] = -3.0e38f;
    l_i[tid] = 0.0f;
  }

  v8f o0 = (v8f){0.f,0.f,0.f,0.f,0.f,0.f,0.f,0.f};
  v8f o1 = o0;
  const int ml    = lane & 15;
  const int kh    = (lane >> 4) * 2;
  const int rbase = (lane >> 4) * 8;

  stage_wait();
  __syncthreads();

  for (int t = 0; t < SDIM / 64; ++t) {
    // -- stage K / V tiles + key norms ------------------------------------
    {
      const size_t kbase = ((size_t)bb * SDIM + t * 64) * EDIM + hh * DH;
#pragma unroll
      for (int rr = 0; rr < 64; rr += 16) {
        stage16(&Ks[rr + r][c4], &K[kbase + (size_t)(rr + r) * EDIM + c4]);
        stage16(&Vs[rr + r][c4], &V[kbase + (size_t)(rr + r) * EDIM + c4]);
      }
    }
    if (tid < 64)
      kn2s[tid] = fminf(KN2[((size_t)bb * SDIM + t * 64 + tid) * HDIM + hh],
                        1.0f - 1e-5f);
    stage_wait();
    __syncthreads();

    // -- raw qk^T for this wave's 16x32 slab (batched fragments) ----------
    v8f sc0 = (v8f){0.f,0.f,0.f,0.f,0.f,0.f,0.f,0.f};
    v8f sc1 = sc0;
    {
      v2f aq[8], bk0[8], bk1[8];
#pragma unroll
      for (int k8 = 0; k8 < 8; ++k8) {
        const int kk = 4 * k8 + kh;
        aq[k8].x  = Qs[wm * 16 + ml][kk];
        aq[k8].y  = Qs[wm * 16 + ml][kk + 1];
        bk0[k8].x = Ks[wn * 32 + ml][kk];
        bk0[k8].y = Ks[wn * 32 + ml][kk + 1];
        bk1[k8].x = Ks[wn * 32 + 16 + ml][kk];
        bk1[k8].y = Ks[wn * 32 + 16 + ml][kk + 1];
      }
#pragma unroll
      for (int k8 = 0; k8 < 8; ++k8) {
        sc0 = wmma_f32(aq[k8], bk0[k8], sc0);
        sc1 = wmma_f32(aq[k8], bk1[k8], sc1);
      }
    }

    // -- hyperbolic distance transform + per-row max ----------------------
    const float eps = 1e-5f;
    const float kn0 = kn2s[wn * 32 + ml];
    const float kn1 = kn2s[wn * 32 + 16 + ml];
#pragma unroll
    for (int i = 0; i < 8; ++i) {
      const float qn = qn2s[wm * 16 + rbase + i];
      float d0 = 2.0f * (sc0[i] - qn * kn0) / ((1.0f - qn) * (1.0f - kn0) + eps);
      float d1 = 2.0f * (sc1[i] - qn * kn1) / ((1.0f - qn) * (1.0f - kn1) + eps);
      sc0[i] = -sqrtf(fmaxf(d0, eps)) * 0.125f;   // / sqrt(Dh)
      sc1[i] = -sqrtf(fmaxf(d1, eps)) * 0.125f;
      float v = fmaxf(sc0[i], sc1[i]);
#pragma unroll
      for (int d = 1; d < 16; d <<= 1) v = fmaxf(v, __shfl_xor(v, d, 32));
      if (ml == 0) pred[wn][wm * 16 + rbase + i] = v;
    }
    __syncthreads();
    if (tid < 64) {
      float mn = fmaxf(m_i[tid], fmaxf(pred[0][tid], pred[1][tid]));
      fsc[tid] = __expf(m_i[tid] - mn);
      m_i[tid] = mn;
    }
    __syncthreads();

    // -- rescale O, materialize P in LDS, partial row sums ----------------
#pragma unroll
    for (int i = 0; i < 8; ++i) {
      const int row = wm * 16 + rbase + i;
      const float mn = m_i[row];
      const float f  = fsc[row];
      o0[i] *= f;
      o1[i] *= f;
      float p0 = __expf(sc0[i] - mn);
      float p1 = __expf(sc1[i] - mn);
      Ps[row][wn * 32 + ml]      = p0;
      Ps[row][wn * 32 + 16 + ml] = p1;
      float v = p0 + p1;
#pragma unroll
      for (int d = 1; d < 16; d <<= 1) v += __shfl_xor(v, d, 32);
      if (ml == 0) pred[wn][row] = v;
    }
    __syncthreads();
    if (tid < 64) l_i[tid] = l_i[tid] * fsc[tid] + pred[0][tid] + pred[1][tid];

    // -- O += P * V (WMMA, K = 64 keys, two batched half-chunks) ----------
#pragma unroll
    for (int half = 0; half < 2; ++half) {
      v2f ap[8], bv0[8], bv1[8];
#pragma unroll
      for (int k8 = 0; k8 < 8; ++k8) {
        const int kk = half * 32 + 4 * k8 + kh;
        ap[k8].x  = Ps[wm * 16 + ml][kk];
        ap[k8].y  = Ps[wm * 16 + ml][kk + 1];
        bv0[k8].x = Vs[kk][wn * 32 + ml];
        bv0[k8].y = Vs[kk + 1][wn * 32 + ml];
        bv1[k8].x = Vs[kk][wn * 32 + 16 + ml];
        bv1[k8].y = Vs[kk + 1][wn * 32 + 16 + ml];
      }
#pragma unroll
      for (int k8 = 0; k8 < 8; ++k8) {
        o0 = wmma_f32(ap[k8], bv0[k8], o0);
        o1 = wmma_f32(ap[k8], bv1[k8], o1);
      }
    }
    __syncthreads();
  }

  // -- normalize and write ------------------------------------------------
#pragma unroll
  for (int i = 0; i < 8; ++i) {
    const int row = wm * 16 + rbase + i;
    const float inv = 1.0f / l_i[row];
    const size_t off = rowbase + (size_t)row * EDIM;
    O[off + wn * 32 + ml]      = o0[i] * inv;
    O[off + wn * 32 + 16 + ml] = o1[i] * inv;
  }
}

// ---------------------------------------------------------------------------
extern "C" void kernel_launch(void* const* d_in, const int* in_sizes, int n_in,
                              void* d_out, int out_size, void* d_ws,
                              size_t ws_size, hipStream_t stream) {
  (void)in_sizes; (void)n_in; (void)out_size; (void)ws_size;
  const float* x  = (const float*)d_in[0];
  const float* Wq = (const float*)d_in[1];
  const float* bq = (const float*)d_in[2];
  const float* Wk = (const float*)d_in[3];
  const float* bk = (const float*)d_in[4];
  const float* Wv = (const float*)d_in[5];
  const float* bv = (const float*)d_in[6];
  const float* Wo = (const float*)d_in[7];
  const float* bo = (const float*)d_in[8];
  float* out = (float*)d_out;

  float* ws  = (float*)d_ws;
  float* mq  = ws;                                   // [8192,512]
  float* mk  = mq + (size_t)MROWS * EDIM;
  float* mv  = mk + (size_t)MROWS * EDIM;
  float* ao  = mv + (size_t)MROWS * EDIM;            // attention output
  float* qn2 = ao + (size_t)MROWS * EDIM;            // [8192,8]
  float* kn2 = qn2 + (size_t)MROWS * HDIM;
  float* mo  = mq;                                   // reuse after attention

  const dim3 gg(EDIM / 64, MROWS / 64);              // 8 x 128 blocks

  // Q/K/V projections: mx = x @ W^T
  gemm_nt_kernel<<<gg, 128, 0, stream>>>(x, Wq, mq);
  gemm_nt_kernel<<<gg, 128, 0, stream>>>(x, Wk, mk);
  gemm_nt_kernel<<<gg, 128, 0, stream>>>(x, Wv, mv);

  // hyperbolic linear (mobius_matvec scale + project + mobius_add + project)
  const int hypBlocks = MROWS / 8;                   // 8 waves / block
  hyp_pointwise_kernel<<<hypBlocks, 256, 0, stream>>>(x, mq, bq, mq);
  hyp_pointwise_kernel<<<hypBlocks, 256, 0, stream>>>(x, mk, bk, mk);
  hyp_pointwise_kernel<<<hypBlocks, 256, 0, stream>>>(x, mv, bv, mv);

  // per-head squared norms of q and k
  head_norms_kernel<<<(MROWS * HDIM) / 8, 256, 0, stream>>>(mq, qn2);
  head_norms_kernel<<<(MROWS * HDIM) / 8, 256, 0, stream>>>(mk, kn2);

  // flash hyperbolic attention
  flash_attn_kernel<<<dim3(SDIM / 64, HDIM, BDIM), 256, 0, stream>>>(
      mq, mk, mv, qn2, kn2, ao);

  // output projection + hyperbolic linear -> final output
  gemm_nt_kernel<<<gg, 128, 0, stream>>>(ao, Wo, mo);
  hyp_pointwise_kernel<<<hypBlocks, 256, 0, stream>>>(ao, mo, bo, out);
}